// RnntLoss_75282186764755
// MI455X (gfx1250) — compile-verified
//
#include <hip/hip_runtime.h>
#include <stdint.h>

// Problem constants (from reference setup_inputs)
#define RN  8
#define RT  256
#define RU  128
#define RU1 129
#define RC  512

// ---------------- CDNA5 async global->LDS helpers (gfx1250) ----------------
// Per-lane DMA of 16B global -> LDS, tracked by ASYNCcnt (CDNA5 ISA).
// th:TH_LOAD_NT — activations are streamed exactly once; keep them out of L2 so
// the phase-1 outputs (reused by phase 2) stay resident in the 192MB L2.
__device__ __forceinline__ void async_b128_nt(uint32_t lds_off, const void* gaddr) {
  asm volatile("global_load_async_to_lds_b128 %0, %1, off th:TH_LOAD_NT"
               :: "v"(lds_off), "v"(gaddr) : "memory");
}
__device__ __forceinline__ void wait_async_le4() {
  asm volatile("s_wait_asynccnt 0x4" ::: "memory");
}
__device__ __forceinline__ void wait_async_0() {
  asm volatile("s_wait_asynccnt 0x0" ::: "memory");
}

// ---------------------------------------------------------------------------
// Phase 1: per-row logsumexp over C=512; emit blank_lp and emit_lp.
// One wave (32 lanes) per row; row streamed into LDS via async b128 loads,
// double-buffered. 2KB/row, 541MB total -> HBM-bandwidth bound (~23us floor).
// ---------------------------------------------------------------------------
__global__ __launch_bounds__(256)
void rnnt_phase1(const float* __restrict__ acts, const int* __restrict__ targets,
                 float* __restrict__ blank_out, float* __restrict__ emit_out) {
  __shared__ float lds[8 * 2 * RC];                 // 8 waves * 2 buffers * 512 f32 = 32KB
  const int lane = threadIdx.x & 31;
  const int wave = threadIdx.x >> 5;
  const int wpb  = blockDim.x >> 5;
  const long long totalRows = (long long)RN * RT * RU1;
  long long row = (long long)blockIdx.x * wpb + wave;
  const long long rstride = (long long)gridDim.x * wpb;

  float* wbuf = &lds[wave * 2 * RC];
  const uint32_t wbase = (uint32_t)(uintptr_t)wbuf; // low 32 bits of generic ptr = LDS offset

  if (row >= totalRows) return;                     // wave-uniform; no barriers in kernel

  auto issue = [&](long long r, int b) {
    const float* g = acts + r * (long long)RC + lane * 4;
    uint32_t l = wbase + (uint32_t)b * (RC * 4) + (uint32_t)lane * 16;
#pragma unroll
    for (int j = 0; j < 4; ++j)                     // 4 x (32 lanes x 16B) = 2KB row
      async_b128_nt(l + (uint32_t)j * 512, g + j * 128);
  };

  issue(row, 0);
  int buf = 0;
  for (;;) {
    long long nxt = row + rstride;
    bool has = (nxt < totalRows);
    if (has) { issue(nxt, buf ^ 1); wait_async_le4(); }  // buf ready (in-order completion)
    else     { wait_async_0(); }

    const float4* lb4 = (const float4*)(wbuf + buf * RC);
    float4 v[4];
#pragma unroll
    for (int j = 0; j < 4; ++j) v[j] = lb4[lane + j * 32];

    float m = -3.4e38f;
#pragma unroll
    for (int j = 0; j < 4; ++j)
      m = fmaxf(m, fmaxf(fmaxf(v[j].x, v[j].y), fmaxf(v[j].z, v[j].w)));
#pragma unroll
    for (int off = 16; off >= 1; off >>= 1) m = fmaxf(m, __shfl_xor(m, off, 32));

    float s = 0.f;
#pragma unroll
    for (int j = 0; j < 4; ++j)
      s += __expf(v[j].x - m) + __expf(v[j].y - m) +
           __expf(v[j].z - m) + __expf(v[j].w - m);
#pragma unroll
    for (int off = 16; off >= 1; off >>= 1) s += __shfl_xor(s, off, 32);

    if (lane == 0) {
      float lse = m + __logf(s);
      const float* lb = wbuf + buf * RC;
      int u = (int)(row % RU1);
      long long nt = row / RU1;
      int t = (int)(nt % RT);
      int n = (int)(nt / RT);
      blank_out[row] = lb[0] - lse;                 // lp[t,u,BLANK]
      if (u < RU) {
        int tg = targets[n * RU + u];               // 1 <= tg < C
        emit_out[((long long)n * RT + t) * RU + u] = lb[tg] - lse;  // lp[t,u,tgt[u]]
      }
    }
    if (!has) break;
    row = nxt; buf ^= 1;
  }
}

// ---------------------------------------------------------------------------
// Phase 2: alpha forward DP, anti-diagonal wavefront. One workgroup/sample.
// alpha[t][u] = logaddexp(alpha[t-1][u]+blank[t-1][u], alpha[t][u-1]+emit[t][u-1])
// Both dependencies live on diagonal d-1 -> two ping-pong diagonals in LDS.
// ---------------------------------------------------------------------------
__global__ __launch_bounds__(256)
void rnnt_phase2(const float* __restrict__ blank, const float* __restrict__ emit,
                 const int* __restrict__ alen, const int* __restrict__ tlen,
                 float* __restrict__ losses) {
  __shared__ float dA[RU1], dB[RU1];
  __shared__ float ll_s;
  const int n  = blockIdx.x;
  const int Tn = alen[n];
  const int Un = tlen[n];
  const float* bl = blank + (long long)n * RT * RU1;
  const float* em = emit  + (long long)n * RT * RU;
  const int j = threadIdx.x;

  if (j == 0) {
    dA[0] = 0.0f;                                   // alpha[0][0]
    if (Tn == 1 && Un == 0) ll_s = bl[0];
  }
  __syncthreads();

  float* prev = dA;
  float* cur  = dB;
  for (int d = 1; d <= (RT - 1) + RU; ++d) {
    if (j <= RU && j <= d) {
      int t = d - j;
      if (t >= 0 && t < RT) {
        float val;
        if (t == 0) {                               // row0 cumsum of emit[0]
          val = prev[j - 1] + em[j - 1];
        } else if (j == 0) {                        // first column: blanks only
          val = prev[0] + bl[(t - 1) * RU1];
        } else {
          float a  = prev[j]     + bl[(t - 1) * RU1 + j];
          float b  = prev[j - 1] + em[t * RU + (j - 1)];
          float mx = fmaxf(a, b);
          val = mx + log1pf(__expf(fminf(a, b) - mx));
        }
        cur[j] = val;
        if (t == Tn - 1 && j == Un)                 // capture alpha[Tn-1][Un]
          ll_s = val + bl[(Tn - 1) * RU1 + Un];
      }
    }
    __syncthreads();
    float* tmp = prev; prev = cur; cur = tmp;
  }
  if (j == 0) losses[n] = -ll_s;
}

// ---------------------------------------------------------------------------
// Phase 3: mean of per-sample losses.
// ---------------------------------------------------------------------------
__global__ void rnnt_phase3(const float* __restrict__ losses, float* __restrict__ out) {
  if (threadIdx.x == 0 && blockIdx.x == 0) {
    float s = 0.f;
#pragma unroll
    for (int i = 0; i < RN; ++i) s += losses[i];
    out[0] = s * (1.0f / RN);
  }
}

extern "C" void kernel_launch(void* const* d_in, const int* in_sizes, int n_in,
                              void* d_out, int out_size, void* d_ws, size_t ws_size,
                              hipStream_t stream) {
  const float* acts    = (const float*)d_in[0];
  const int*   targets = (const int*)d_in[1];
  const int*   alen    = (const int*)d_in[2];
  const int*   tlen    = (const int*)d_in[3];

  float* blank  = (float*)d_ws;                         // N*T*(U+1) f32
  float* emit   = blank + (size_t)RN * RT * RU1;        // N*T*U     f32
  float* losses = emit  + (size_t)RN * RT * RU;         // N         f32  (~2.1MB total)

  rnnt_phase1<<<2048, 256, 0, stream>>>(acts, targets, blank, emit);
  rnnt_phase2<<<RN, 256, 0, stream>>>(blank, emit, alen, tlen, losses);
  rnnt_phase3<<<1, 32, 0, stream>>>(losses, (float*)d_out);

  (void)in_sizes; (void)n_in; (void)out_size; (void)ws_size;
}